// ContextualModel_75806172774985
// MI455X (gfx1250) — compile-verified
//
#include <hip/hip_runtime.h>

// CDNA5 / gfx1250, wave32. Memory-bound streaming kernel (~336 MB read + 64 MB
// write -> ~17 us floor @ 23.3 TB/s). Math folded into one GEMM:
//   ys = (mask*q0*feats)[B,16] x Wcomb^T[16,4],  Wcomb = W_reg @ W_kernel
// executed as a chain of 4x V_WMMA_F32_16X16X4_F32 per 16-batch group.
// D tile repacked through a 256B per-wave LDS slot into one coalesced
// global_store_b64 per group. Main loop is guard-free (partial group peeled).

typedef float v2f __attribute__((ext_vector_type(2)));
typedef float v8f __attribute__((ext_vector_type(8)));

#define DIM_M 4
#define DIM_Q 5
#define DIM_C 4

template<bool GUARDED>
__device__ __forceinline__ void do_group(
    int bbase, int B, int n, int hi, int f0, int lane, float* slotp,
    const float* __restrict__ xss, const int* __restrict__ seq_lengths,
    float* __restrict__ out, v2f bmat)
{
    const int b  = bbase + n;
    const int bc = GUARDED ? ((b < B) ? b : (B - 1)) : b;  // junk rows never stored
    const float* xb = xss + (size_t)bc * (DIM_M * DIM_Q);
    const int len   = seq_lengths[bc];

    v8f acc = {0.f, 0.f, 0.f, 0.f, 0.f, 0.f, 0.f, 0.f};

    // K=16 contraction as 4 chained 16x16x4 f32 WMMAs (s = WMMA index).
    #pragma unroll
    for (int s = 0; s < DIM_M; ++s) {
        const float q0 = xb[s * DIM_Q + 0];
        const float sc = (s < len) ? q0 : 0.f;   // fold mask * q0 into A
        v2f a;
        a.x = sc * xb[s * DIM_Q + 1 + f0];
        a.y = sc * xb[s * DIM_Q + 2 + f0];
        acc = __builtin_amdgcn_wmma_f32_16x16x4_f32(
            false, a, false, bmat, (short)0, acc, false, false);
    }

    // D layout: acc[r] @ lane(n,hi) = ys[bbase + r + 8*hi, n], valid for n<4.
    // Repack 64 valid dwords into contiguous LDS, then one 256B coalesced store.
    if (n < DIM_M) {
        #pragma unroll
        for (int r = 0; r < 8; ++r)
            slotp[(r + 8 * hi) * DIM_M + n] = acc[r];
    }
    __builtin_amdgcn_wave_barrier();   // same-wave DS ops are in-order; pin order
    const v2f packed = *(const v2f*)&slotp[lane * 2];
    __builtin_amdgcn_wave_barrier();   // keep load before next iteration's stores

    if (!GUARDED) {
        *(v2f*)(out + (size_t)bbase * DIM_M + lane * 2) = packed;
    } else {
        const int i0 = bbase * DIM_M + lane * 2;
        const int lim = B * DIM_M;
        if (i0 < lim)     out[i0]     = packed.x;
        if (i0 + 1 < lim) out[i0 + 1] = packed.y;
    }
}

__global__ __launch_bounds__(256) void contextual_wmma_kernel(
    const float* __restrict__ xss,          // [B, 4, 5]
    const int*   __restrict__ seq_lengths,  // [B]
    const float* __restrict__ W_kernel,     // [4, 4]  (C, Q-1)
    const float* __restrict__ W_reg,        // [4, 4]  (M, C)
    float*       __restrict__ out,          // [B, 4]
    int B)
{
    __shared__ float repack[8][64];         // one 256B D-repack slot per wave

    const int lane = (int)(threadIdx.x & 31);
    const int n    = lane & 15;             // batch row within 16-row A/D tile
    const int hi   = lane >> 4;             // K/f pair {0,1} vs {2,3}
    const int f0   = 2 * hi;
    float* slotp   = repack[threadIdx.x >> 5];

    // ---- Loop-invariant B matrix (4x16 f32 = 2 VGPRs/lane) ----
    // B[f, j] = Wcomb[j, f] for j<4 else 0;  Wcomb[m,f] = sum_c Wr[m,c]*Wk[c,f]
    const int m = n & 3;
    float w0 = 0.f, w1 = 0.f;
    #pragma unroll
    for (int c = 0; c < DIM_C; ++c) {
        const float wr = W_reg[m * DIM_C + c];
        w0 += wr * W_kernel[c * (DIM_Q - 1) + f0];
        w1 += wr * W_kernel[c * (DIM_Q - 1) + f0 + 1];
    }
    v2f bmat;
    bmat.x = (n < DIM_M) ? w0 : 0.f;
    bmat.y = (n < DIM_M) ? w1 : 0.f;

    // Wave-uniform loop control in SGPRs (readfirstlane scalarizes).
    const int waveId = __builtin_amdgcn_readfirstlane(
        (int)((blockIdx.x * blockDim.x + threadIdx.x) >> 5));
    const int nWaves = __builtin_amdgcn_readfirstlane(
        (int)((gridDim.x * blockDim.x) >> 5));
    const int nFull  = B >> 4;              // full 16-batch groups
    const int rem    = B & 15;

    for (int g = waveId; g < nFull; g += nWaves) {
        const int bbase = g << 4;

        // Speculative prefetch of this wave's next group into GL2 (clamped).
        long pb = (long)bbase + n + (long)nWaves * 16;
        if (pb >= (long)B) pb = (long)B - 1;
        __builtin_prefetch(xss + pb * (DIM_M * DIM_Q), 0, 1);

        do_group<false>(bbase, B, n, hi, f0, lane, slotp,
                        xss, seq_lengths, out, bmat);
    }

    // At most one partial group (B % 16 != 0 only); owned by one wave.
    if (rem && waveId == (nFull % nWaves)) {
        do_group<true>(nFull << 4, B, n, hi, f0, lane, slotp,
                       xss, seq_lengths, out, bmat);
    }
}

extern "C" void kernel_launch(void* const* d_in, const int* in_sizes, int n_in,
                              void* d_out, int out_size, void* d_ws, size_t ws_size,
                              hipStream_t stream) {
    const float* xss = (const float*)d_in[0];
    const int*   seq = (const int*)d_in[1];
    const float* Wk  = (const float*)d_in[2];
    const float* Wr  = (const float*)d_in[3];
    float*       out = (float*)d_out;
    const int B = in_sizes[1];   // seq_lengths element count == batch count

    dim3 block(256);             // 8 wave32 waves per workgroup
    dim3 grid(4096);             // grid-stride over the 250K 16-batch groups
    hipLaunchKernelGGL(contextual_wmma_kernel, grid, block, 0, stream,
                       xss, seq, Wk, Wr, out, B);
}